// ParallelBeamReconstruction2D_80281528696888
// MI455X (gfx1250) — compile-verified
//
#include <hip/hip_runtime.h>
#include <math.h>

#define B_ 16
#define A_ 720
#define D_ 1024
#define H_ 512
#define W_ 512
#define PI_ 3.14159265358979323846
#define KC 128   // K-chunk staged in LDS per block

typedef float v2f __attribute__((ext_vector_type(2)));
typedef float v8f __attribute__((ext_vector_type(8)));
typedef unsigned int u32;
typedef u32 u32x4 __attribute__((ext_vector_type(4)));
typedef int  i32x4 __attribute__((ext_vector_type(4)));
typedef int  i32x8 __attribute__((ext_vector_type(8)));

#if defined(__has_builtin)
#if __has_builtin(__builtin_amdgcn_tensor_load_to_lds)
#define HAVE_TDM 1
#endif
#endif

#ifdef HAVE_TDM
// Issue one TDM 2D tile load: tile_h rows x tile_w elems (f32), global row
// stride row_stride elems, into contiguous LDS at lds_ptr. Wave-level op;
// caller must be a single wave and then s_wait_tensorcnt + block barrier.
// D# packing per CDNA5 ISA.
__device__ __forceinline__ void tdm_load_tile_f32(void* lds_ptr, const void* gptr,
                                                  u32 tile_w, u32 tile_h,
                                                  u32 row_stride) {
    unsigned long long ga = (unsigned long long)(uintptr_t)gptr;
    u32 lds_off = (u32)(uintptr_t)lds_ptr;   // flat LDS addr: low 32 bits = LDS offset
    u32x4 g0;
    g0.x = 1u;                                   // count=1, user mode, no gather
    g0.y = lds_off;                              // lds_addr (bytes)
    g0.z = (u32)(ga & 0xFFFFFFFFu);              // global_addr[31:0]
    g0.w = (u32)((ga >> 32) & 0x01FFFFFFu) | (2u << 30);  // global_addr[56:32] | type=2
    i32x8 g1;
    g1[0] = (int)(2u << 16);                     // data_size=2 (4B); mask=0; no flags
    g1[1] = (int)0xFFFF0000u;                    // tensor_dim0[15:0]=0xFFFF (td0=0x7FFFFFFF)
    g1[2] = (int)0xFFFF7FFFu;                    // td0[31:16]=0x7FFF | td1[15:0]=0xFFFF
    g1[3] = (int)((tile_w << 16) | 0x7FFFu);     // td1[31:16]=0x7FFF | tile_dim0
    g1[4] = (int)(tile_h & 0xFFFFu);             // tile_dim1 | tile_dim2=0
    g1[5] = (int)row_stride;                     // tensor_dim0_stride[31:0]
    g1[6] = 0;                                   // stride[47:32] | tensor_dim1_stride lo
    g1[7] = 0;                                   // tensor_dim1_stride hi
    i32x4 gz = {0, 0, 0, 0};
#if __clang_major__ >= 23
    i32x8 gz8 = {0, 0, 0, 0, 0, 0, 0, 0};
    __builtin_amdgcn_tensor_load_to_lds(g0, g1, gz, gz, gz8, 0);
#else
    __builtin_amdgcn_tensor_load_to_lds(g0, g1, gz, gz, 0);
#endif
}
#endif

// ---------------------------------------------------------------- angle table
__global__ void k_angles(float2* __restrict__ ang) {
    int a = blockIdx.x * blockDim.x + threadIdx.x;
    if (a < A_) {
        float th = (float)a * (float)(PI_ / A_);
        float s, c;
        __sincosf(th, &s, &c);
        ang[a] = make_float2(c, s);
    }
}

// ------------------------------------------------- h[m] = IDFT(filt)[m].real
__global__ void k_build_h(const float* __restrict__ filt, float* __restrict__ h) {
    int m = blockIdx.x * blockDim.x + threadIdx.x;
    if (m >= D_) return;
    const float step = (float)(2.0 * PI_ / D_);
    float acc = 0.0f;
    for (int f = 0; f < D_; ++f) {
        int p = (f * m) & (D_ - 1);               // exact phase reduction
        acc = fmaf(filt[f], __cosf((float)p * step), acc);
    }
    h[m] = acc * (1.0f / (float)D_);
}

// ----------------------------------------------- Ht[n][k] = h[(n - k) mod D]
__global__ void k_build_Ht(const float* __restrict__ h, float* __restrict__ Ht) {
    int i = blockIdx.x * blockDim.x + threadIdx.x;
    if (i >= D_ * D_) return;
    int n = i >> 10;
    int k = i & (D_ - 1);
    Ht[i] = h[(n - k) & (D_ - 1)];
}

// ------------------------------------------------------- filtering as a GEMM
// Y[M,N] = X[M,K] * H[K,N], M = B*A = 11520, N = K = D = 1024 (fp32 WMMA).
// Block = 8 waves sharing one 16-row M-tile; A chunk (16 x KC f32, 8KB) staged
// into LDS by the Tensor Data Mover, fragments read back via ds_load_b64.
// B fragments stream from Ht (H transposed) as contiguous float2 per lane.
__global__ void __launch_bounds__(256)
k_gemm_wmma(const float* __restrict__ X, const float* __restrict__ Ht,
            float* __restrict__ Y) {
    __shared__ float As[16 * KC];                 // staged A chunk
    const int lane = threadIdx.x & 31;
    const int wave = threadIdx.x >> 5;
    const int mt   = blockIdx.x >> 3;             // 720 M-tiles
    const int nt   = ((blockIdx.x & 7) << 3) + wave;  // 64 N-tiles
    const int row0 = mt * 16;
    const int n0   = nt * 16;
    const int half = lane >> 4;
    const int m    = lane & 15;

    const float* bp      = Ht + (size_t)(n0 + m) * D_ + 2 * half;
    const float* As_frag = As + m * KC + 2 * half;

    v8f c = {0.f, 0.f, 0.f, 0.f, 0.f, 0.f, 0.f, 0.f};
    for (int k0 = 0; k0 < D_; k0 += KC) {
#ifdef HAVE_TDM
        if (threadIdx.x < 32) {                   // one wave issues the TDM op
            tdm_load_tile_f32(As, X + (size_t)row0 * D_ + k0, KC, 16, D_);
            __builtin_amdgcn_s_wait_tensorcnt(0);
        }
#else
        for (int i = threadIdx.x; i < 16 * KC; i += 256) {
            int r = i >> 7, kk = i & (KC - 1);
            As[i] = X[(size_t)(row0 + r) * D_ + k0 + kk];
        }
#endif
        __syncthreads();
        __builtin_prefetch(bp + k0 + KC, 0, 0);   // global_prefetch_b8 on B stream
        #pragma unroll 8
        for (int kk = 0; kk < KC; kk += 4) {
            v2f a = *(const v2f*)(As_frag + kk);  // ds_load_b64
            v2f b = *(const v2f*)(bp + k0 + kk);  // global_load_b64
            c = __builtin_amdgcn_wmma_f32_16x16x4_f32(
                    false, a, false, b, (short)0, c, false, false);
        }
        __syncthreads();
    }

    // C/D layout: VGPR r -> row M = r + 8*half, column N = lane&15
    float* yp = Y + (size_t)(row0 + 8 * half) * D_ + n0 + m;
    #pragma unroll
    for (int r = 0; r < 8; ++r)
        yp[(size_t)r * D_] = c[r];
}

// ------------------------------------ transpose [B][A][D] -> [A][D][B] (LDS)
__global__ void k_transpose(const float* __restrict__ Y, float* __restrict__ Yt) {
    __shared__ float tile[16][17];
    int a  = blockIdx.x;
    int d0 = blockIdx.y * 16;
    int tx = threadIdx.x & 15;
    int ty = threadIdx.x >> 4;
    tile[tx][ty] = Y[((size_t)ty * A_ + a) * D_ + d0 + tx];   // coalesced read
    __syncthreads();
    Yt[((size_t)a * D_ + d0 + ty) * B_ + tx] = tile[ty][tx];  // coalesced write
}

// ---------------------------------------------------------- backprojection
__device__ inline float4 fma4(float4 acc, float w0, float4 q, float w1, float4 r) {
    acc.x = fmaf(w0, q.x, fmaf(w1, r.x, acc.x));
    acc.y = fmaf(w0, q.y, fmaf(w1, r.y, acc.y));
    acc.z = fmaf(w0, q.z, fmaf(w1, r.z, acc.z));
    acc.w = fmaf(w0, q.w, fmaf(w1, r.w, acc.w));
    return acc;
}

__global__ void __launch_bounds__(256)
k_backproject(const float* __restrict__ Yt, const float2* __restrict__ ang,
              float* __restrict__ out) {
    __shared__ float2 angs[A_];                   // 5.76 KB angle cache
    for (int i = threadIdx.x; i < A_; i += 256) angs[i] = ang[i];
    __syncthreads();

    const int xx = blockIdx.x * 16 + (threadIdx.x & 15);
    const int yy = blockIdx.y * 16 + (threadIdx.x >> 4);
    const float px = (float)xx - 0.5f * (float)(W_ - 1);
    const float py = (float)yy - 0.5f * (float)(H_ - 1);
    const float det_center = 0.5f * (float)(D_ - 1);

    float4 acc0 = {0, 0, 0, 0}, acc1 = {0, 0, 0, 0};
    float4 acc2 = {0, 0, 0, 0}, acc3 = {0, 0, 0, 0};

    #pragma unroll 2
    for (int a = 0; a < A_; ++a) {
        float2 cs = angs[a];
        float t  = fmaf(px, cs.x, fmaf(py, cs.y, det_center));
        float fi = floorf(t);
        int   i0 = (int)fi;
        float w  = t - fi;
        bool valid = (i0 >= 0) && (i0 < D_ - 1);
        int i0c = i0 < 0 ? 0 : (i0 > D_ - 1 ? D_ - 1 : i0);
        int i1c = i0c + 1 > D_ - 1 ? D_ - 1 : i0c + 1;
        float w1 = valid ? w : 0.0f;
        float w0 = valid ? 1.0f - w : 0.0f;

        const float4* p0 = (const float4*)(Yt + ((size_t)a * D_ + i0c) * B_);
        const float4* p1 = (const float4*)(Yt + ((size_t)a * D_ + i1c) * B_);
        acc0 = fma4(acc0, w0, p0[0], w1, p1[0]);
        acc1 = fma4(acc1, w0, p0[1], w1, p1[1]);
        acc2 = fma4(acc2, w0, p0[2], w1, p1[2]);
        acc3 = fma4(acc3, w0, p0[3], w1, p1[3]);
    }

    const float sc = (float)(PI_ / A_);
    const size_t HW = (size_t)H_ * W_;
    float* o = out + (size_t)yy * W_ + xx;
    o[0  * HW] = acc0.x * sc;  o[1  * HW] = acc0.y * sc;
    o[2  * HW] = acc0.z * sc;  o[3  * HW] = acc0.w * sc;
    o[4  * HW] = acc1.x * sc;  o[5  * HW] = acc1.y * sc;
    o[6  * HW] = acc1.z * sc;  o[7  * HW] = acc1.w * sc;
    o[8  * HW] = acc2.x * sc;  o[9  * HW] = acc2.y * sc;
    o[10 * HW] = acc2.z * sc;  o[11 * HW] = acc2.w * sc;
    o[12 * HW] = acc3.x * sc;  o[13 * HW] = acc3.y * sc;
    o[14 * HW] = acc3.z * sc;  o[15 * HW] = acc3.w * sc;
}

// --------------------------------------------------------------------- host
extern "C" void kernel_launch(void* const* d_in, const int* in_sizes, int n_in,
                              void* d_out, int out_size, void* d_ws, size_t ws_size,
                              hipStream_t stream) {
    (void)in_sizes; (void)n_in; (void)out_size; (void)ws_size;
    const float* sino = (const float*)d_in[0];   // [B, A, D] f32
    const float* filt = (const float*)d_in[1];   // [D] f32
    float* out = (float*)d_out;                  // [B, H, W] f32

    char*  ws  = (char*)d_ws;
    size_t off = 0;
    auto alloc = [&](size_t bytes) -> void* {
        void* p = ws + off;
        off += (bytes + 255) & ~(size_t)255;
        return p;
    };
    float*  h   = (float*)alloc((size_t)D_ * sizeof(float));
    float*  Ht  = (float*)alloc((size_t)D_ * D_ * sizeof(float));
    float*  Y   = (float*)alloc((size_t)B_ * A_ * D_ * sizeof(float));
    float*  Yt  = (float*)alloc((size_t)A_ * D_ * B_ * sizeof(float));
    float2* ang = (float2*)alloc((size_t)A_ * sizeof(float2));

    k_angles  <<<(A_ + 255) / 256, 256, 0, stream>>>(ang);
    k_build_h <<<D_ / 256, 256, 0, stream>>>(filt, h);
    k_build_Ht<<<(D_ * D_) / 256, 256, 0, stream>>>(h, Ht);

    // (B*A/16) M-tiles x (D/16) N-tiles; 8 waves (N-tiles) per block
    k_gemm_wmma<<<46080 / 8, 256, 0, stream>>>(sino, Ht, Y);

    k_transpose<<<dim3(A_, D_ / 16), 256, 0, stream>>>(Y, Yt);

    k_backproject<<<dim3(W_ / 16, H_ / 16), 256, 0, stream>>>(Yt, ang, out);
}